// mLSTMCell_45887430590546
// MI455X (gfx1250) — compile-verified
//
#include <hip/hip_runtime.h>
#include <math.h>

typedef float v2f __attribute__((ext_vector_type(2)));
typedef float v4f __attribute__((ext_vector_type(4)));
typedef float v8f __attribute__((ext_vector_type(8)));

#define B_SZ 128
#define DDIM 1024

// ---------------------------------------------------------------------------
// GEMM: out[m][n] = act( scale * sum_k A[m][k] * W[n][k] + bias[n] )
// i.e. A @ W^T, A:(M=128 x K), W:(N x K), both row-major.
// Uses V_WMMA_F32_16X16X4_F32. grid = (N/64, M/16), block = 128 (4 waves).
// A-frag (16x4 f32): lane<16 holds A[m0+l][k0..k0+1], lane>=16 holds A[m0+l][k0+2..3]
// B-frag (4x16 f32): lane<16 holds W[n0+l][k0..k0+1], lane>=16 holds W[n0+l][k0+2..3]
// D (16x16 f32): VGPR r -> row m0 + r + 8*(lane>=16), col n0 + (lane&15)
// ---------------------------------------------------------------------------
__global__ __launch_bounds__(128)
void gemm_xwt_wmma(const float* __restrict__ A, const float* __restrict__ W,
                   const float* __restrict__ bias, float* __restrict__ out,
                   float scale, int act_sigmoid, int K, int N)
{
    const int wave = threadIdx.x >> 5;
    const int lane = threadIdx.x & 31;
    const int half = lane >> 4;
    const int l16  = lane & 15;
    const int n0 = (blockIdx.x * 4 + wave) * 16;
    const int m0 = blockIdx.y * 16;

    const float* arow = A + (size_t)(m0 + l16) * K + half * 2;
    const float* wrow = W + (size_t)(n0 + l16) * K + half * 2;

    v8f acc = {0.f, 0.f, 0.f, 0.f, 0.f, 0.f, 0.f, 0.f};
#pragma unroll 4
    for (int k0 = 0; k0 < K; k0 += 4) {
        v2f av = *(const v2f*)(arow + k0);
        v2f bv = *(const v2f*)(wrow + k0);
        acc = __builtin_amdgcn_wmma_f32_16x16x4_f32(false, av, false, bv,
                                                    (short)0, acc, false, false);
    }

    const int ncol = n0 + l16;
    const float bb = bias ? bias[ncol] : 0.0f;
#pragma unroll
    for (int r = 0; r < 8; ++r) {
        float y = acc[r] * scale + bb;
        if (act_sigmoid) y = 1.0f / (1.0f + __expf(-y));
        out[(size_t)(m0 + half * 8 + r) * N + ncol] = y;
    }
}

// ---------------------------------------------------------------------------
// Gates: z_i = x[b]·w_i + b_i ; z_f = x[b]·w_f + b_f ; stabilized exp gates.
// One wave per batch row.
// ---------------------------------------------------------------------------
__global__ __launch_bounds__(32)
void gates_kernel(const float* __restrict__ x, const float* __restrict__ wi,
                  const float* __restrict__ wib, const float* __restrict__ wf,
                  const float* __restrict__ wfb, const float* __restrict__ m_prev,
                  float* __restrict__ ip, float* __restrict__ fp,
                  float* __restrict__ m_out)
{
    const int b = blockIdx.x;
    const int lane = threadIdx.x;
    const v4f* x4  = (const v4f*)(x + (size_t)b * DDIM);
    const v4f* wi4 = (const v4f*)wi;
    const v4f* wf4 = (const v4f*)wf;
    float si = 0.f, sf = 0.f;
#pragma unroll
    for (int t = 0; t < DDIM / 128; ++t) {
        int j = lane + t * 32;
        v4f xv = x4[j], iv = wi4[j], fv = wf4[j];
        si += xv.x * iv.x + xv.y * iv.y + xv.z * iv.z + xv.w * iv.w;
        sf += xv.x * fv.x + xv.y * fv.y + xv.z * fv.z + xv.w * fv.w;
    }
    for (int off = 16; off; off >>= 1) {
        si += __shfl_xor(si, off, 32);
        sf += __shfl_xor(sf, off, 32);
    }
    if (lane == 0) {
        float zi = si + wib[0];
        float zf = sf + wfb[0];
        float mp = m_prev[b];
        float mt = fmaxf(zf + mp, zi);
        ip[b] = __expf(zi - mt);
        fp[b] = __expf(zf + mp - mt);
        m_out[b] = mt;
    }
}

// ---------------------------------------------------------------------------
// The 1 GB streaming kernel: C_t = f*C_prev + i*(v outer k), fused with
// Cq[b][i] = sum_j C_t[b][i][j] * q[b][j].
// One wave per C row (4 KB); NT 128-bit loads/stores for the C stream,
// cached loads for k/q (reused 1024x per batch row -> stay in L2/WGP$).
// grid = B*DDIM/8 blocks of 256 threads (8 waves).
// ---------------------------------------------------------------------------
__global__ __launch_bounds__(256)
void update_kernel(const float* __restrict__ C_prev, const float* __restrict__ kt,
                   const float* __restrict__ qt, const float* __restrict__ vt,
                   const float* __restrict__ ip, const float* __restrict__ fp,
                   float* __restrict__ C_out, float* __restrict__ Cq)
{
    const int wave = threadIdx.x >> 5;
    const int lane = threadIdx.x & 31;
    const long long row = (long long)blockIdx.x * 8 + wave;  // b*DDIM + i
    const int b = (int)(row >> 10);
    const int i = (int)(row & (DDIM - 1));

    const float f  = fp[b];
    const float iv = ip[b] * vt[(size_t)b * DDIM + i];

    const v4f* cp = (const v4f*)(C_prev + row * DDIM);
    v4f*       co = (v4f*)(C_out + row * DDIM);
    const v4f* k4 = (const v4f*)(kt + (size_t)b * DDIM);
    const v4f* q4 = (const v4f*)(qt + (size_t)b * DDIM);

    float acc = 0.f;
#pragma unroll
    for (int t = 0; t < DDIM / 128; ++t) {
        int j = lane + t * 32;
        v4f c  = __builtin_nontemporal_load(cp + j);
        v4f kv = k4[j];
        v4f qv = q4[j];
        v4f cn = f * c + iv * kv;
        __builtin_nontemporal_store(cn, co + j);
        acc += cn.x * qv.x + cn.y * qv.y + cn.z * qv.z + cn.w * qv.w;
    }
    for (int off = 16; off; off >>= 1) acc += __shfl_xor(acc, off, 32);
    if (lane == 0) Cq[row] = acc;
}

// ---------------------------------------------------------------------------
// n_t = f*n_prev + i*k ; denom[b] = max(|n_t · q|, 1). One wave per row.
// ---------------------------------------------------------------------------
__global__ __launch_bounds__(32)
void nq_kernel(const float* __restrict__ n_prev, const float* __restrict__ kt,
               const float* __restrict__ qt, const float* __restrict__ ip,
               const float* __restrict__ fp, float* __restrict__ n_out,
               float* __restrict__ denom)
{
    const int b = blockIdx.x;
    const int lane = threadIdx.x;
    const float f = fp[b], ii = ip[b];
    const v4f* np = (const v4f*)(n_prev + (size_t)b * DDIM);
    const v4f* k4 = (const v4f*)(kt + (size_t)b * DDIM);
    const v4f* q4 = (const v4f*)(qt + (size_t)b * DDIM);
    v4f*       no = (v4f*)(n_out + (size_t)b * DDIM);
    float acc = 0.f;
#pragma unroll
    for (int t = 0; t < DDIM / 128; ++t) {
        int j = lane + t * 32;
        v4f nn = f * np[j] + ii * k4[j];
        no[j] = nn;
        v4f qv = q4[j];
        acc += nn.x * qv.x + nn.y * qv.y + nn.z * qv.z + nn.w * qv.w;
    }
    for (int off = 16; off; off >>= 1) acc += __shfl_xor(acc, off, 32);
    if (lane == 0) denom[b] = fmaxf(fabsf(acc), 1.0f);
}

// h_head = o * Cq / denom  (elementwise over B*DDIM)
__global__ __launch_bounds__(256)
void hhead_kernel(const float* __restrict__ o, const float* __restrict__ Cq,
                  const float* __restrict__ denom, float* __restrict__ hh)
{
    int idx = blockIdx.x * 256 + threadIdx.x;  // 0 .. B*DDIM-1
    int b = idx >> 10;
    hh[idx] = o[idx] * Cq[idx] / denom[b];
}

// ---------------------------------------------------------------------------
extern "C" void kernel_launch(void* const* d_in, const int* in_sizes, int n_in,
                              void* d_out, int out_size, void* d_ws, size_t ws_size,
                              hipStream_t stream)
{
    const float* x    = (const float*)d_in[0];
    const float* Cp   = (const float*)d_in[1];
    const float* np   = (const float*)d_in[2];
    const float* mp   = (const float*)d_in[3];
    const float* Wq   = (const float*)d_in[4];
    const float* Wk   = (const float*)d_in[5];
    const float* Wv   = (const float*)d_in[6];
    const float* wiw  = (const float*)d_in[7];
    const float* wib  = (const float*)d_in[8];
    const float* wfw  = (const float*)d_in[9];
    const float* wfb  = (const float*)d_in[10];
    const float* Wo   = (const float*)d_in[11];
    const float* Wob  = (const float*)d_in[12];
    const float* Wout = (const float*)d_in[13];

    const size_t BD = (size_t)B_SZ * DDIM;          // 131072
    float* out   = (float*)d_out;
    float* h_out = out;                              // (B, D)
    float* C_out = out + BD;                         // (B, D, D)
    float* n_out = C_out + (size_t)B_SZ * DDIM * DDIM;
    float* m_out = n_out + BD;

    float* ws  = (float*)d_ws;
    float* q   = ws;                                 // BD
    float* k   = q   + BD;                           // BD
    float* v   = k   + BD;                           // BD
    float* o   = v   + BD;                           // BD  (sigmoid applied)
    float* ipb = o   + BD;                           // B
    float* fpb = ipb + B_SZ;                         // B
    float* Cq  = fpb + B_SZ;                         // BD
    float* den = Cq  + BD;                           // B
    float* hh  = den + B_SZ;                         // BD

    const float scale = 0.03125f;                    // 1/sqrt(1024)
    dim3 ggrid(DDIM / 64, B_SZ / 16);                // (16, 8), 4 waves/block

    gemm_xwt_wmma<<<ggrid, 128, 0, stream>>>(x, Wq, nullptr, q, scale, 0, DDIM, DDIM);
    gemm_xwt_wmma<<<ggrid, 128, 0, stream>>>(x, Wk, nullptr, k, scale, 0, DDIM, DDIM);
    gemm_xwt_wmma<<<ggrid, 128, 0, stream>>>(x, Wv, nullptr, v, 1.0f, 0, DDIM, DDIM);
    gemm_xwt_wmma<<<ggrid, 128, 0, stream>>>(x, Wo, Wob, o, 1.0f, 1, DDIM, DDIM);

    gates_kernel<<<B_SZ, 32, 0, stream>>>(x, wiw, wib, wfw, wfb, mp, ipb, fpb, m_out);

    update_kernel<<<(int)(BD / 8), 256, 0, stream>>>(Cp, k, q, v, ipb, fpb, C_out, Cq);

    nq_kernel<<<B_SZ, 32, 0, stream>>>(np, k, q, ipb, fpb, n_out, den);

    hhead_kernel<<<(int)(BD / 256), 256, 0, stream>>>(o, Cq, den, hh);

    gemm_xwt_wmma<<<ggrid, 128, 0, stream>>>(hh, Wout, nullptr, h_out, 1.0f, 0, DDIM, DDIM);
}